// GCN_40750649704955
// MI455X (gfx1250) — compile-verified
//
#include <hip/hip_runtime.h>

typedef __attribute__((ext_vector_type(2))) float v2f;
typedef __attribute__((ext_vector_type(8))) float v8f;

static constexpr int K_IN = 128;   // input feature width for both layers' GEMMs

// Hardware fp32 atomic add (global_atomic_add_f32), relaxed / device scope.
__device__ __forceinline__ void atomAddF(float* p, float v) {
    __hip_atomic_fetch_add(p, v, __ATOMIC_RELAXED, __HIP_MEMORY_SCOPE_AGENT);
}

// ---------------------------------------------------------------------------
// Degree / normalization
// ---------------------------------------------------------------------------
__global__ __launch_bounds__(256) void gcn_deg_count(const int* __restrict__ dst,
                                                     float* __restrict__ deg, int E) {
    const int e = blockIdx.x * 256 + threadIdx.x;
    if (e < E) atomAddF(&deg[dst[e]], 1.0f);
}

__global__ __launch_bounds__(256) void gcn_deg_to_dinv(float* __restrict__ deg, int N) {
    const int i = blockIdx.x * 256 + threadIdx.x;
    if (i < N) deg[i] = rsqrtf(deg[i] + 1.0f);   // +1 = self loop; always > 0
}

// ---------------------------------------------------------------------------
// fp32 WMMA GEMM:  C[M x NOUT] = A[M x 128] @ W[128 x NOUT]
// Block = 256 threads = 8 waves. W staged in LDS in a K-pair-interleaved
// layout -- element (k, col) lives at lW[2*((k>>1)*NOUT + col) + (k&1)] -- so
// each V_WMMA_F32_16X16X4_F32 B-fragment {W[k][c], W[k+1][c]} is one aligned
// ds_load_b64 (no register shuffles). Staging uses the gfx1250 async
// global->LDS path (global_load_async_to_lds_b32 + s_wait_asynccnt): linear
// coalesced global reads, packed per-lane LDS scatter for free.
// Each wave: 16 x NOUT slab, A-fragment reused across NOUT/16 tiles. Uniform
// control flow: A row index clamped (not predicated) so EXEC stays all-ones
// around the WMMAs; out-of-range rows masked only at the store.
// ---------------------------------------------------------------------------
template <int NOUT>
__global__ __launch_bounds__(256) void gcn_gemm_wmma(const float* __restrict__ A,
                                                     const float* __restrict__ W,
                                                     float* __restrict__ C, int M) {
    __shared__ alignas(16) float lW[K_IN * NOUT];
    const int tid = threadIdx.x;
    for (int i = tid; i < K_IN * NOUT; i += 256) {
        const int k   = i / NOUT;   // NOUT is a power of two -> shifts
        const int col = i % NOUT;
        const int pk  = 2 * ((k >> 1) * NOUT + col) + (k & 1);
#if __has_builtin(__builtin_amdgcn_global_load_async_to_lds_b32) && \
    __has_builtin(__builtin_amdgcn_s_wait_asynccnt)
        __builtin_amdgcn_global_load_async_to_lds_b32(
            (__attribute__((address_space(1))) int*)(const_cast<float*>(W) + i),
            (__attribute__((address_space(3))) int*)(&lW[pk]), 0, 0);
#else
        lW[pk] = W[i];
#endif
    }
#if __has_builtin(__builtin_amdgcn_global_load_async_to_lds_b32) && \
    __has_builtin(__builtin_amdgcn_s_wait_asynccnt)
    __builtin_amdgcn_s_wait_asynccnt(0);
#endif
    __syncthreads();

    const int wave  = tid >> 5;
    const int lane  = tid & 31;
    const int l16   = lane & 15;
    const int kHalf = (lane >> 4) << 1;            // 0 (lanes 0-15) or 2 (lanes 16-31)
    const int mBase = blockIdx.x * 128 + wave * 16;
    const int row   = mBase + l16;
    const int rowC  = row < M ? row : (M - 1);     // clamped; masked at store
    const float* arow = A + (size_t)rowC * K_IN;
    __builtin_prefetch(arow, 0, 3);
    __builtin_prefetch(arow + 64, 0, 3);

    constexpr int NT = NOUT / 16;
    v8f acc[NT] = {};

    for (int k0 = 0; k0 < K_IN; k0 += 4) {
        // A fragment, 16x4 f32: lane l16 = row, kHalf selects K pair {0,1}/{2,3}
        const v2f a = *reinterpret_cast<const v2f*>(arow + k0 + kHalf);
        const int p = (k0 + kHalf) >> 1;           // interleaved K-pair index
#pragma unroll
        for (int t = 0; t < NT; ++t) {
            // B fragment: both K components contiguous in packed LDS -> one b64
            const v2f b = *reinterpret_cast<const v2f*>(
                &lW[2 * (p * NOUT + t * 16 + l16)]);
            acc[t] = __builtin_amdgcn_wmma_f32_16x16x4_f32(
                false, a, false, b, (short)0, acc[t], false, false);
        }
    }

    // C/D layout: VGPR r holds M=r (lanes 0-15) / M=r+8 (lanes 16-31), N = l16
    const int mOff = (lane >> 4) * 8;
#pragma unroll
    for (int t = 0; t < NT; ++t) {
#pragma unroll
        for (int r = 0; r < 8; ++r) {
            const int m = mBase + mOff + r;
            if (m < M) C[(size_t)m * NOUT + t * 16 + l16] = acc[t][r];
        }
    }
}

// ---------------------------------------------------------------------------
// Edge-parallel scatter-add: one wave per edge, F/32 contiguous floats per lane
// (coalesced 512B-per-wave gathers; fp32 atomics resolve in the 192MB L2).
// ---------------------------------------------------------------------------
template <int F>
__global__ __launch_bounds__(256) void gcn_scatter(const float* __restrict__ h,
                                                   const int* __restrict__ src,
                                                   const int* __restrict__ dst,
                                                   const float* __restrict__ dinv,
                                                   float* __restrict__ out, int E) {
    const int gw   = blockIdx.x * 8 + (threadIdx.x >> 5);
    const int lane = threadIdx.x & 31;
    if (gw >= E) return;
    const int s = src[gw];
    const int d = dst[gw];
    const float norm = dinv[s] * dinv[d];
    const float* hs = h + (size_t)s * F;
    float* od = out + (size_t)d * F;
    constexpr int PL = F / 32;
    const int base = lane * PL;
    if constexpr (PL == 4) {
        const float4 v = *reinterpret_cast<const float4*>(hs + base);
        atomAddF(od + base + 0, v.x * norm);
        atomAddF(od + base + 1, v.y * norm);
        atomAddF(od + base + 2, v.z * norm);
        atomAddF(od + base + 3, v.w * norm);
    } else {
        const float2 v = *reinterpret_cast<const float2*>(hs + base);
        atomAddF(od + base + 0, v.x * norm);
        atomAddF(od + base + 1, v.y * norm);
    }
}

// ---------------------------------------------------------------------------
// Finalize: agg += self-loop term h[i]*dinv^2, add bias, optional ReLU.
// Runs after the scatter kernel (stream-ordered) so no atomics needed.
// ---------------------------------------------------------------------------
template <int F, bool RELU>
__global__ __launch_bounds__(256) void gcn_finalize(float* __restrict__ agg,
                                                    const float* __restrict__ h,
                                                    const float* __restrict__ dinv,
                                                    const float* __restrict__ bias,
                                                    int N) {
    const size_t i = (size_t)blockIdx.x * 256 + threadIdx.x;
    if (i >= (size_t)N * F) return;
    const int n = (int)(i / F);
    const int f = (int)(i % F);
    const float dv = dinv[n];
    float v = agg[i] + h[i] * (dv * dv) + bias[f];
    if (RELU) v = fmaxf(v, 0.0f);
    agg[i] = v;
}

// ---------------------------------------------------------------------------
extern "C" void kernel_launch(void* const* d_in, const int* in_sizes, int n_in,
                              void* d_out, int out_size, void* d_ws, size_t ws_size,
                              hipStream_t stream) {
    (void)n_in; (void)out_size; (void)ws_size;
    const float* x  = (const float*)d_in[0];
    const int*   ei = (const int*)d_in[1];
    const float* W1 = (const float*)d_in[2];
    const float* b1 = (const float*)d_in[3];
    const float* W2 = (const float*)d_in[4];
    const float* b2 = (const float*)d_in[5];
    float* out = (float*)d_out;

    const int N = in_sizes[0] / 128;
    const int E = in_sizes[1] / 2;
    const int* src = ei;        // edge_index[0]
    const int* dst = ei + E;    // edge_index[1]

    // Workspace layout (256B aligned): deg/dinv | h1 | agg1 | h2
    char* ws = (char*)d_ws;
    size_t off = 0;
    auto wsAlloc = [&](size_t bytes) -> void* {
        void* p = ws + off;
        off += (bytes + 255) & ~size_t(255);
        return p;
    };
    float* dinv = (float*)wsAlloc((size_t)N * 4);
    float* h1   = (float*)wsAlloc((size_t)N * 128 * 4);
    float* agg1 = (float*)wsAlloc((size_t)N * 128 * 4);
    float* h2   = (float*)wsAlloc((size_t)N * 64 * 4);

    (void)hipMemsetAsync(dinv, 0, (size_t)N * 4, stream);
    (void)hipMemsetAsync(agg1, 0, (size_t)N * 128 * 4, stream);
    (void)hipMemsetAsync(out,  0, (size_t)N * 64 * 4, stream);

    // Normalization
    gcn_deg_count<<<(E + 255) / 256, 256, 0, stream>>>(dst, dinv, E);
    gcn_deg_to_dinv<<<(N + 255) / 256, 256, 0, stream>>>(dinv, N);

    // Layer 1: GEMM -> scatter -> (self-loop + bias + ReLU)
    gcn_gemm_wmma<128><<<(N + 127) / 128, 256, 0, stream>>>(x, W1, h1, N);
    gcn_scatter<128><<<(E + 7) / 8, 256, 0, stream>>>(h1, src, dst, dinv, agg1, E);
    {
        const size_t total = (size_t)N * 128;
        gcn_finalize<128, true><<<(int)((total + 255) / 256), 256, 0, stream>>>(
            agg1, h1, dinv, b1, N);
    }

    // Layer 2: GEMM -> scatter -> (self-loop + bias)
    gcn_gemm_wmma<64><<<(N + 127) / 128, 256, 0, stream>>>(agg1, W2, h2, N);
    gcn_scatter<64><<<(E + 7) / 8, 256, 0, stream>>>(h2, src, dst, dinv, out, E);
    {
        const size_t total = (size_t)N * 64;
        gcn_finalize<64, false><<<(int)((total + 255) / 256), 256, 0, stream>>>(
            out, h2, dinv, b2, N);
    }
}